// Aggregator_15212774163210
// MI455X (gfx1250) — compile-verified
//
#include <hip/hip_runtime.h>

#define CH 64

typedef __attribute__((ext_vector_type(2))) float v2f;
typedef __attribute__((ext_vector_type(8))) float v8f;

// ---------------------------------------------------------------------------
// out[M, 64] = X[M,64] @ (scale * B[64,64])   via V_WMMA_F32_16X16X4_F32
// One wave per 16-row block; 8 waves (256 threads) per CTA -> 128 rows/CTA.
// ---------------------------------------------------------------------------
__global__ __launch_bounds__(256)
void gemm64_wmma_kernel(const float* __restrict__ X,
                        const float* __restrict__ Bmat,
                        float* __restrict__ out,
                        int M, float scale) {
    __shared__ float lb[CH * CH];           // 16 KB: aspect_emb, pre-scaled
    for (int i = threadIdx.x; i < CH * CH; i += 256)
        lb[i] = scale * Bmat[i];
    __syncthreads();

    const int wave = threadIdx.x >> 5;
    const int lane = threadIdx.x & 31;
    const int n    = lane & 15;   // A: row M=n ; B/C/D: column n
    const int kh   = lane >> 4;   // A/B: K-half ; C/D: M-half
    const int row0 = blockIdx.x * 128 + wave * 16;

    // A-matrix fragments for all 16 K-steps (16x4 f32 layout, ISA 7.12.2)
    v2f a[16];
    {
        const float* xrow  = X + (size_t)(row0 + n) * CH;
        const bool   valid = (row0 + n) < M;
#pragma unroll
        for (int ks = 0; ks < 16; ++ks) {
            if (valid) {
                a[ks].x = xrow[4 * ks + 2 * kh + 0];
                a[ks].y = xrow[4 * ks + 2 * kh + 1];
            } else {
                a[ks].x = 0.0f;
                a[ks].y = 0.0f;
            }
        }
    }

#pragma unroll
    for (int nt = 0; nt < 4; ++nt) {        // four 16-wide N tiles
        v8f c = {};
#pragma unroll
        for (int ks = 0; ks < 16; ++ks) {   // K = 64 in steps of 4
            const int kb = 4 * ks + 2 * kh;
            v2f b;
            b.x = lb[(kb + 0) * CH + nt * 16 + n];
            b.y = lb[(kb + 1) * CH + nt * 16 + n];
            c = __builtin_amdgcn_wmma_f32_16x16x4_f32(
                    /*neg_a=*/false, a[ks], /*neg_b=*/false, b,
                    /*c_mod=*/(short)0, c, /*reuse_a=*/false, /*reuse_b=*/false);
        }
        // C/D layout: VGPR r -> M = r (lanes 0-15) / r+8 (lanes 16-31)
#pragma unroll
        for (int r = 0; r < 8; ++r) {
            const int row = row0 + r + 8 * kh;
            if (row < M)
                out[(size_t)row * CH + nt * 16 + n] = c[r];
        }
    }
}

// ---------------------------------------------------------------------------
// Zero fill
// ---------------------------------------------------------------------------
__global__ __launch_bounds__(256)
void zero_kernel(float* __restrict__ p, long long n) {
    long long i = (long long)blockIdx.x * blockDim.x + threadIdx.x;
    if (i < n) p[i] = 0.0f;
}

// ---------------------------------------------------------------------------
// Edge scatter: one wave (32 lanes) per edge, 2 channels per lane.
// sums[head] += entity_emb[tail] * weight[etype-2];  counts[head] += 1
// ---------------------------------------------------------------------------
__global__ __launch_bounds__(256)
void edge_scatter_kernel(const float* __restrict__ entity_emb,
                         const float* __restrict__ weight,
                         const int* __restrict__ head,
                         const int* __restrict__ tail,
                         const int* __restrict__ etype,
                         float* __restrict__ sums,
                         float* __restrict__ counts,
                         int E) {
    long long gid = (long long)blockIdx.x * blockDim.x + threadIdx.x;
    int e   = (int)(gid >> 5);
    int ch2 = (int)(gid & 31);          // channel pair 0..31
    if (e >= E) return;

    int h = head[e];
    int t = tail[e];
    int r = etype[e] - 2;

    const float2* te = (const float2*)(entity_emb + (size_t)t * CH);
    const float2* wr = (const float2*)(weight + (size_t)r * CH);
    float2 ev = te[ch2];
    float2 wv = wr[ch2];

    float* dst = sums + (size_t)h * CH + ch2 * 2;
    atomicAdd(dst + 0, ev.x * wv.x);
    atomicAdd(dst + 1, ev.y * wv.y);
    if (ch2 == 0) atomicAdd(counts + h, 1.0f);
}

// ---------------------------------------------------------------------------
// In-place scatter-mean normalization: sums[i*64+c] /= max(counts[i], 1)
// ---------------------------------------------------------------------------
__global__ __launch_bounds__(256)
void normalize_kernel(float* __restrict__ sums,
                      const float* __restrict__ counts,
                      long long nTotal) {
    long long i = (long long)blockIdx.x * blockDim.x + threadIdx.x;
    if (i < nTotal) {
        float cnt = counts[i >> 6];
        sums[i] = sums[i] / fmaxf(cnt, 1.0f);
    }
}

// ---------------------------------------------------------------------------
extern "C" void kernel_launch(void* const* d_in, const int* in_sizes, int n_in,
                              void* d_out, int out_size, void* d_ws, size_t ws_size,
                              hipStream_t stream) {
    const float* entity_emb = (const float*)d_in[0];
    const float* item_emb   = (const float*)d_in[1]; // unused: softmax row-sum == 1
    const float* user_emb   = (const float*)d_in[2]; // unused: softmax row-sum == 1
    const float* aspect_emb = (const float*)d_in[3];
    const int*   edge_index = (const int*)d_in[4];   // [2, E] flat
    const int*   edge_type  = (const int*)d_in[5];
    const float* ua_mat     = (const float*)d_in[6];
    const float* ia_mat     = (const float*)d_in[7];
    const float* weight     = (const float*)d_in[8];
    (void)item_emb; (void)user_emb; (void)n_in; (void)out_size;

    const int n_entities = in_sizes[0] / CH;
    const int n_items    = in_sizes[1] / CH;
    const int n_users    = in_sizes[2] / CH;
    const int n_edges    = in_sizes[5];

    // d_out = [item_agg | entity_agg | user_agg], flat f32
    float* out_item   = (float*)d_out;
    float* out_entity = out_item   + (size_t)n_items    * CH;
    float* out_user   = out_entity + (size_t)n_entities * CH;
    float* counts     = (float*)d_ws;        // n_entities floats of scratch

    const int* head = edge_index;
    const int* tail = edge_index + n_edges;

    // ---- entity_agg: zero -> scatter -> mean ----
    {
        long long nEnt = (long long)n_entities * CH;
        zero_kernel<<<(int)((nEnt + 255) / 256), 256, 0, stream>>>(out_entity, nEnt);
        zero_kernel<<<(n_entities + 255) / 256, 256, 0, stream>>>(counts, n_entities);

        long long threads = (long long)n_edges * 32;
        edge_scatter_kernel<<<(int)((threads + 255) / 256), 256, 0, stream>>>(
            entity_emb, weight, head, tail, edge_type, out_entity, counts, n_edges);

        normalize_kernel<<<(int)((nEnt + 255) / 256), 256, 0, stream>>>(
            out_entity, counts, nEnt);
    }

    // ---- item_agg = 2 * ia_mat @ aspect_emb ----
    gemm64_wmma_kernel<<<(n_items + 127) / 128, 256, 0, stream>>>(
        ia_mat, aspect_emb, out_item, n_items, 2.0f);

    // ---- user_agg = 2 * ua_mat @ aspect_emb ----
    gemm64_wmma_kernel<<<(n_users + 127) / 128, 256, 0, stream>>>(
        ua_mat, aspect_emb, out_user, n_users, 2.0f);
}